// EncoderLayer_58222576665005
// MI455X (gfx1250) — compile-verified
//
#include <hip/hip_runtime.h>

// ------------------------------------------------------------------
// Problem constants (from reference): S=1024, B=2, D=1024, H=16,
// depth=64, DFF=4096.  M = S*B = 2048 rows, row-major 2048x1024.
// ------------------------------------------------------------------
#define SEQ     1024
#define DMODEL  1024
#define NHEAD   16
#define DEPTH   64
#define DFF     4096
#define MROWS   2048
#define ATTN_SCALE 0.125f          // 1/sqrt(64)
#define NEGMASK   (-1000000000.0f)
#define LN_EPS    1e-5f

typedef float v2f __attribute__((ext_vector_type(2)));
typedef float v8f __attribute__((ext_vector_type(8)));

// D(16x16 f32) = A(16x4 f32) x B(4x16 f32) + C
__device__ __forceinline__ v8f wmma_f32(v2f a, v2f b, v8f c) {
    return __builtin_amdgcn_wmma_f32_16x16x4_f32(false, a, false, b,
                                                 (short)0, c, false, false);
}

// ---------------- K0: per-head column sums of w_qs / w_ks ----------
// wqc[h*1024+d] = sum_n w_qs[d][h*64+n]
__global__ void k_colsum(const float* __restrict__ wq, const float* __restrict__ wk,
                         float* __restrict__ wqc, float* __restrict__ wkc) {
    int idx = blockIdx.x * blockDim.x + threadIdx.x;
    if (idx >= NHEAD * DMODEL) return;
    int h = idx >> 10, d = idx & 1023;
    const float* rq = wq + (size_t)d * DMODEL + h * DEPTH;
    const float* rk = wk + (size_t)d * DMODEL + h * DEPTH;
    float sq = 0.f, sk = 0.f;
#pragma unroll 8
    for (int n = 0; n < DEPTH; ++n) { sq += rq[n]; sk += rk[n]; }
    wqc[h * DMODEL + d] = sq;
    wkc[h * DMODEL + d] = sk;
}

// ---------------- K1: qs / kts reductions --------------------------
// qs[(b*H+h)*S+s]  = scale * (X_row(r) . wqc[h]),  r = b*1024+s
// kts[(b*H+h)*S+s] =         (X_row(r) . wkc[h])
__global__ void __launch_bounds__(256) k_qkred(const float* __restrict__ X,
                                               const float* __restrict__ wqc,
                                               const float* __restrict__ wkc,
                                               float* __restrict__ qs,
                                               float* __restrict__ kts) {
    __shared__ float xrow[DMODEL];
    int r = blockIdx.x;
    const float* xr = X + (size_t)r * DMODEL;
    for (int i = threadIdx.x; i < DMODEL; i += 256) xrow[i] = xr[i];
    __syncthreads();
    int lane = threadIdx.x & 31, w = threadIdx.x >> 5;
    int b = r >> 10, s = r & 1023;
    for (int t4 = 0; t4 < 4; ++t4) {
        int task = w * 4 + t4;                  // 0..31 : 16 q-heads then 16 k-heads
        int h = task & 15;
        const float* col = (task < 16 ? wqc : wkc) + (size_t)h * DMODEL;
        float acc = 0.f;
        for (int d = lane; d < DMODEL; d += 32) acc += xrow[d] * col[d];
        for (int m = 16; m >= 1; m >>= 1) acc += __shfl_xor(acc, m, 32);
        if (lane == 0) {
            int o = (b * NHEAD + h) * SEQ + s;
            if (task < 16) qs[o] = ATTN_SCALE * acc;
            else           kts[o] = acc;
        }
    }
}

// ---------------- GEMM (f32 WMMA 16x16x4): C = op(A@B) -------------
// Block tile 128x64, K-chunk 32. 8 waves, each wave: 16 rows x 64 cols.
template <bool RELU, bool HAS_BIAS, bool HAS_RES>
__global__ void __launch_bounds__(256) k_gemm_f32(const float* __restrict__ A,
                                                  const float* __restrict__ B,
                                                  const float* __restrict__ bias,
                                                  const float* __restrict__ res,
                                                  float* __restrict__ C,
                                                  int Nn, int Kk) {
    __shared__ float Asl[128][33];
    __shared__ float Bsl[32][65];
    const int rowBase = blockIdx.x * 128;
    const int colBase = blockIdx.y * 64;
    const int tid = threadIdx.x, lane = tid & 31, w = tid >> 5;

    const v8f vzero = {0.f,0.f,0.f,0.f,0.f,0.f,0.f,0.f};
    v8f acc[4];
#pragma unroll
    for (int t = 0; t < 4; ++t) acc[t] = vzero;

    const int mrow  = w * 16 + (lane & 15);
    const int koff2 = (lane >> 4) * 2;

    for (int k0 = 0; k0 < Kk; k0 += 32) {
#pragma unroll
        for (int i = 0; i < 4; ++i) {              // A: 128x32
            int vid = tid + i * 256;
            int ar = vid >> 3, ac = (vid & 7) * 4;
            float4 v = *reinterpret_cast<const float4*>(
                A + (size_t)(rowBase + ar) * Kk + k0 + ac);
            Asl[ar][ac + 0] = v.x; Asl[ar][ac + 1] = v.y;
            Asl[ar][ac + 2] = v.z; Asl[ar][ac + 3] = v.w;
        }
#pragma unroll
        for (int i = 0; i < 2; ++i) {              // B: 32x64
            int vid = tid + i * 256;
            int br = vid >> 4, bc = (vid & 15) * 4;
            float4 v = *reinterpret_cast<const float4*>(
                B + (size_t)(k0 + br) * Nn + colBase + bc);
            Bsl[br][bc + 0] = v.x; Bsl[br][bc + 1] = v.y;
            Bsl[br][bc + 2] = v.z; Bsl[br][bc + 3] = v.w;
        }
        __syncthreads();
#pragma unroll
        for (int kk = 0; kk < 32; kk += 4) {
            v2f a;
            a.x = Asl[mrow][kk + koff2];
            a.y = Asl[mrow][kk + koff2 + 1];
#pragma unroll
            for (int t = 0; t < 4; ++t) {
                int n = t * 16 + (lane & 15);
                v2f bb;
                bb.x = Bsl[kk + koff2][n];
                bb.y = Bsl[kk + koff2 + 1][n];
                acc[t] = wmma_f32(a, bb, acc[t]);
            }
        }
        __syncthreads();
    }
    // epilogue: C layout reg i -> row = i + 8*(lane>=16), col = lane&15
    int rbase = rowBase + w * 16 + ((lane >> 4) * 8);
#pragma unroll
    for (int t = 0; t < 4; ++t) {
        int col = colBase + t * 16 + (lane & 15);
        float bv = HAS_BIAS ? bias[col] : 0.f;
#pragma unroll
        for (int i = 0; i < 8; ++i) {
            int row = rbase + i;
            float vv = acc[t][i] + bv;
            if (RELU) vv = fmaxf(vv, 0.f);
            if (HAS_RES) vv += res[(size_t)row * Nn + col];
            C[(size_t)row * Nn + col] = vv;
        }
    }
}

// ---------------- K3: skewed rel_w reductions ----------------------
// adk[row]  = sum_{m<=kk} rel[row][1023-kk+m] * kts[bh][m]
// adkm[row] = same with ktms[m] = kts[m] - 1e9*min(m,64)
__global__ void __launch_bounds__(256) k_relred(const float* __restrict__ rel,
                                                const float* __restrict__ kts,
                                                float* __restrict__ adk,
                                                float* __restrict__ adkm) {
    __shared__ float sh1[256], sh2[256];
    int row = blockIdx.x;                 // (b*H+h)*S + kk
    int kk = row & 1023, bh = row >> 10;
    const float* rrow = rel + (size_t)row * SEQ + (1023 - kk);
    const float* kr = kts + (size_t)bh * SEQ;
    float s1 = 0.f, s2 = 0.f;
    for (int m = threadIdx.x; m <= kk; m += 256) {
        float rv = rrow[m];
        float kv = kr[m];
        int mm = m < 64 ? m : 64;
        s1 += rv * kv;
        s2 += rv * (kv + NEGMASK * (float)mm);
    }
    sh1[threadIdx.x] = s1; sh2[threadIdx.x] = s2;
    __syncthreads();
    for (int st = 128; st >= 1; st >>= 1) {
        if (threadIdx.x < st) {
            sh1[threadIdx.x] += sh1[threadIdx.x + st];
            sh2[threadIdx.x] += sh2[threadIdx.x + st];
        }
        __syncthreads();
    }
    if (threadIdx.x == 0) { adk[row] = sh1[0]; adkm[row] = sh2[0]; }
}

// ---------------- K4: T = sum(adk), u = scale*qs*T -----------------
__global__ void __launch_bounds__(256) k_bmms(const float* __restrict__ adk,
                                              const float* __restrict__ qs,
                                              float* __restrict__ u) {
    __shared__ float sh[256];
    int bh = blockIdx.x;
    float s = 0.f;
    for (int k = threadIdx.x; k < SEQ; k += 256) s += adk[bh * SEQ + k];
    sh[threadIdx.x] = s;
    __syncthreads();
    for (int st = 128; st >= 1; st >>= 1) {
        if (threadIdx.x < st) sh[threadIdx.x] += sh[threadIdx.x + st];
        __syncthreads();
    }
    float T = sh[0];
    for (int s2 = threadIdx.x; s2 < SEQ; s2 += 256)
        u[bh * SEQ + s2] = ATTN_SCALE * qs[bh * SEQ + s2] * T;
}

// ---------------- K5: softmax(u*wvec) @ V_head  (WMMA) -------------
// Grid: (bh=32, sblock=16). Block computes 64 rows x 64 cols of out4.
__global__ void __launch_bounds__(256) k_attn(const float* __restrict__ u,
                                              const float* __restrict__ wvec,
                                              const float* __restrict__ V,
                                              float* __restrict__ out4) {
    __shared__ float wv[SEQ];
    __shared__ float Vl[32][65];
    __shared__ float uloc[64], mrow[64], izrow[64];
    int bh = blockIdx.x;
    int b = bh >> 4, h = bh & 15;
    int s0 = blockIdx.y * 64;
    int tid = threadIdx.x, lane = tid & 31, w = tid >> 5;

    for (int i = tid; i < SEQ; i += 256) wv[i] = wvec[bh * SEQ + i];
    if (tid < 64) uloc[tid] = u[bh * SEQ + s0 + tid];
    __syncthreads();

    // phase 1: per-row max / 1/sum  (wave w owns rows w*8 .. w*8+7)
    for (int rr = 0; rr < 8; ++rr) {
        int rl = w * 8 + rr;
        float us = uloc[rl];
        float mx = -3.4e38f;
        for (int k = lane; k < SEQ; k += 32) mx = fmaxf(mx, us * wv[k]);
        for (int m = 16; m >= 1; m >>= 1) mx = fmaxf(mx, __shfl_xor(mx, m, 32));
        float sum = 0.f;
        for (int k = lane; k < SEQ; k += 32) sum += __expf(us * wv[k] - mx);
        for (int m = 16; m >= 1; m >>= 1) sum += __shfl_xor(sum, m, 32);
        if (lane == 0) { mrow[rl] = mx; izrow[rl] = 1.0f / sum; }
    }
    __syncthreads();

    // phase 2: P @ V_head with WMMA; wave -> 16 rows x 32 cols
    int rb = (w >> 1) * 16;
    int cb = (w & 1) * 32;
    const v8f vzero = {0.f,0.f,0.f,0.f,0.f,0.f,0.f,0.f};
    v8f acc[2]; acc[0] = vzero; acc[1] = vzero;
    int mloc = rb + (lane & 15);
    float us = uloc[mloc], mx = mrow[mloc], iz = izrow[mloc];
    int koff2 = (lane >> 4) * 2;
    const float* Vbase = V + (size_t)(b * SEQ) * DMODEL + h * DEPTH;

    for (int k0 = 0; k0 < SEQ; k0 += 32) {
#pragma unroll
        for (int i = 0; i < 2; ++i) {          // V chunk 32 x 64 -> LDS
            int vid = tid + i * 256;
            int vr = vid >> 4, vc = (vid & 15) * 4;
            float4 t = *reinterpret_cast<const float4*>(
                Vbase + (size_t)(k0 + vr) * DMODEL + vc);
            Vl[vr][vc + 0] = t.x; Vl[vr][vc + 1] = t.y;
            Vl[vr][vc + 2] = t.z; Vl[vr][vc + 3] = t.w;
        }
        __syncthreads();
#pragma unroll
        for (int kk = 0; kk < 32; kk += 4) {
            int kg = k0 + kk + koff2;
            v2f a;
            a.x = __expf(us * wv[kg]     - mx) * iz;
            a.y = __expf(us * wv[kg + 1] - mx) * iz;
#pragma unroll
            for (int t = 0; t < 2; ++t) {
                int n = cb + t * 16 + (lane & 15);
                v2f bb;
                bb.x = Vl[kk + koff2][n];
                bb.y = Vl[kk + koff2 + 1][n];
                acc[t] = wmma_f32(a, bb, acc[t]);
            }
        }
        __syncthreads();
    }
    float* obase = out4 + (size_t)bh * SEQ * DEPTH;
    int rgl = s0 + rb + ((lane >> 4) * 8);
#pragma unroll
    for (int t = 0; t < 2; ++t) {
        int n = cb + t * 16 + (lane & 15);
#pragma unroll
        for (int i = 0; i < 8; ++i)
            obase[(size_t)(rgl + i) * DEPTH + n] = acc[t][i];
    }
}

// ---------------- LayerNorm: out = LN(base [+ add]) * g + b --------
__global__ void __launch_bounds__(256) k_ln(const float* __restrict__ base,
                                            const float* __restrict__ add,
                                            const float* __restrict__ g,
                                            const float* __restrict__ bb,
                                            float* __restrict__ out) {
    __shared__ float row[DMODEL];
    __shared__ float red[256];
    int r = blockIdx.x;
    float s = 0.f;
    for (int i = threadIdx.x; i < DMODEL; i += 256) {
        float v = base[(size_t)r * DMODEL + i] + (add ? add[(size_t)r * DMODEL + i] : 0.f);
        row[i] = v; s += v;
    }
    red[threadIdx.x] = s;
    __syncthreads();
    for (int st = 128; st >= 1; st >>= 1) {
        if (threadIdx.x < st) red[threadIdx.x] += red[threadIdx.x + st];
        __syncthreads();
    }
    float mu = red[0] * (1.0f / DMODEL);
    __syncthreads();
    float s2 = 0.f;
    for (int i = threadIdx.x; i < DMODEL; i += 256) {
        float d = row[i] - mu; s2 += d * d;
    }
    red[threadIdx.x] = s2;
    __syncthreads();
    for (int st = 128; st >= 1; st >>= 1) {
        if (threadIdx.x < st) red[threadIdx.x] += red[threadIdx.x + st];
        __syncthreads();
    }
    float inv = rsqrtf(red[0] * (1.0f / DMODEL) + LN_EPS);
    for (int i = threadIdx.x; i < DMODEL; i += 256)
        out[(size_t)r * DMODEL + i] = (row[i] - mu) * inv * g[i] + bb[i];
}

// ------------------------------------------------------------------
extern "C" void kernel_launch(void* const* d_in, const int* in_sizes, int n_in,
                              void* d_out, int out_size, void* d_ws, size_t ws_size,
                              hipStream_t stream) {
    (void)in_sizes; (void)n_in; (void)out_size; (void)ws_size;
    const float* x     = (const float*)d_in[0];
    const float* w_qs  = (const float*)d_in[1];
    const float* w_ks  = (const float*)d_in[2];
    const float* w_vs  = (const float*)d_in[3];
    const float* rel_w = (const float*)d_in[4];
    const float* w1    = (const float*)d_in[5];
    const float* b1    = (const float*)d_in[6];
    const float* w2    = (const float*)d_in[7];
    const float* b2    = (const float*)d_in[8];
    const float* ln1_g = (const float*)d_in[9];
    const float* ln1_b = (const float*)d_in[10];
    const float* ln2_g = (const float*)d_in[11];
    const float* ln2_b = (const float*)d_in[12];
    float* out = (float*)d_out;

    // workspace layout (floats); aliasing: Y reuses V, G reuses out4
    float* w = (float*)d_ws;
    size_t o = 0;
    float* V    = w + o; o += (size_t)MROWS * DMODEL;   // 2M  (later reused as Y)
    float* qs   = w + o; o += 32768;
    float* kts  = w + o; o += 32768;
    float* adk  = w + o; o += 32768;
    float* adkm = w + o; o += 32768;
    float* u    = w + o; o += 32768;
    float* wqc  = w + o; o += 16384;
    float* wkc  = w + o; o += 16384;
    float* h1   = w + o; o += (size_t)MROWS * DMODEL;   // 2M
    float* R    = w + o; o += (size_t)MROWS * DFF;      // 8M (out4 then G)
    float* out4 = R;
    float* G    = R;
    float* Y    = V;

    k_colsum<<<64, 256, 0, stream>>>(w_qs, w_ks, wqc, wkc);
    k_qkred<<<MROWS, 256, 0, stream>>>(x, wqc, wkc, qs, kts);
    // V = X @ w_vs
    k_gemm_f32<false, false, false><<<dim3(MROWS / 128, DMODEL / 64), 256, 0, stream>>>(
        x, w_vs, nullptr, nullptr, V, DMODEL, DMODEL);
    k_relred<<<2 * NHEAD * SEQ, 256, 0, stream>>>(rel_w, kts, adk, adkm);
    k_bmms<<<2 * NHEAD, 256, 0, stream>>>(adk, qs, u);
    k_attn<<<dim3(2 * NHEAD, SEQ / 64), 256, 0, stream>>>(u, adkm, V, out4);
    // h1 = LN1(x + attn_out)   (attn_out flat == out4 flat)
    k_ln<<<MROWS, 256, 0, stream>>>(x, out4, ln1_g, ln1_b, h1);
    // G = relu(h1 @ w1 + b1)
    k_gemm_f32<true, true, false><<<dim3(MROWS / 128, DFF / 64), 256, 0, stream>>>(
        h1, w1, b1, nullptr, G, DFF, DMODEL);
    // Y = h1 + (G @ w2 + b2)
    k_gemm_f32<false, true, true><<<dim3(MROWS / 128, DMODEL / 64), 256, 0, stream>>>(
        G, w2, b2, h1, Y, DMODEL, DFF);
    // out = LN2(Y)
    k_ln<<<MROWS, 256, 0, stream>>>(Y, nullptr, ln2_g, ln2_b, out);
}